// CrossAttention_8031588843664
// MI455X (gfx1250) — compile-verified
//
#include <hip/hip_runtime.h>
#include <math.h>

typedef float v2f __attribute__((ext_vector_type(2)));
typedef float v8f __attribute__((ext_vector_type(8)));

#define D_MODEL 256
#define NHEAD 8
#define HEAD_DIM 32
#define KNBR 64
#define LN_EPS 1e-5f

// ---------------------------------------------------------------------------
// Generic fp32 WMMA GEMM: C[M,N] = A[M,K] @ B[K,N], all row-major, opt. ReLU.
// Block = 256 threads = 8 waves. Each wave computes a 16x64 C tile (4 WMMA
// accumulators sharing one A fragment). Block tile = 128(M) x 64(N).
// Fragment layouts per CDNA5 ISA 7.12.2:
//   A (16x4):  lane&15 -> M row, K = k0 + (lane>>4)*2 + {vgpr0,vgpr1}
//   B (4x16):  lane&15 -> N col, K = k0 + (lane>>4)*2 + {vgpr0,vgpr1}
//   C (16x16): vgpr i -> M = m0 + i + (lane>>4)*8, N = n0 + (lane&15)
// ---------------------------------------------------------------------------
__global__ __launch_bounds__(256) void gemm_f32_wmma(
    const float* __restrict__ A, const float* __restrict__ B,
    float* __restrict__ C, int M, int N, int K, int relu) {
  const int wave = threadIdx.x >> 5;
  const int lane = threadIdx.x & 31;
  const int m0 = (blockIdx.y * 8 + wave) * 16;
  const int n0 = blockIdx.x * 64;
  if (m0 >= M) return;
  int mrow = m0 + (lane & 15);
  if (mrow >= M) mrow = M - 1;                // clamp loads; stores are guarded
  const int ncol = n0 + (lane & 15);
  const int ksel = (lane >> 4) * 2;
  const float* Arow = A + (size_t)mrow * K;

  v8f c0 = {}, c1 = {}, c2 = {}, c3 = {};
  for (int k0 = 0; k0 < K; k0 += 4) {
    const int ka = k0 + ksel;
    v2f a;
    a.x = Arow[ka];
    a.y = Arow[ka + 1];
    const float* B0 = B + (size_t)ka * N + ncol;
    const float* B1 = B0 + N;
    v2f b0, b1, b2, b3;
    b0.x = B0[0];   b0.y = B1[0];
    b1.x = B0[16];  b1.y = B1[16];
    b2.x = B0[32];  b2.y = B1[32];
    b3.x = B0[48];  b3.y = B1[48];
    c0 = __builtin_amdgcn_wmma_f32_16x16x4_f32(false, a, false, b0, (short)0, c0, false, false);
    c1 = __builtin_amdgcn_wmma_f32_16x16x4_f32(false, a, false, b1, (short)0, c1, false, false);
    c2 = __builtin_amdgcn_wmma_f32_16x16x4_f32(false, a, false, b2, (short)0, c2, false, false);
    c3 = __builtin_amdgcn_wmma_f32_16x16x4_f32(false, a, false, b3, (short)0, c3, false, false);
  }

  const int rbase = m0 + ((lane >> 4) << 3);
  #pragma unroll
  for (int i = 0; i < 8; ++i) {
    const int r = rbase + i;
    if (r < M) {
      float* Crow = C + (size_t)r * N + n0 + (lane & 15);
      float x0 = c0[i], x1 = c1[i], x2 = c2[i], x3 = c3[i];
      if (relu) {
        x0 = fmaxf(x0, 0.f); x1 = fmaxf(x1, 0.f);
        x2 = fmaxf(x2, 0.f); x3 = fmaxf(x3, 0.f);
      }
      Crow[0] = x0; Crow[16] = x1; Crow[32] = x2; Crow[48] = x3;
    }
  }
}

// ---------------------------------------------------------------------------
// Epipolar gathered attention. One wave (32 lanes) per query.
// Phase 1: lane owns keys {lane, lane+32}; fp32 dot over 32 dims per head.
// Softmax per head via wave32 shfl_xor reductions; attn weights -> LDS.
// Phase 2: lane owns output dim (h*32+lane per head), sums over 64 keys.
// ---------------------------------------------------------------------------
__global__ __launch_bounds__(256) void epi_attn(
    const float* __restrict__ q, const float* __restrict__ k,
    const float* __restrict__ v, const int* __restrict__ idx,
    float* __restrict__ msg, int NQ, int L, int S) {
  __shared__ float sq[8][D_MODEL];
  __shared__ float sattn[8][NHEAD * KNBR];
  __shared__ int sidx[8][KNBR];
  const int wave = threadIdx.x >> 5;
  const int lane = threadIdx.x & 31;
  int query = blockIdx.x * 8 + wave;
  const bool active = (query < NQ);
  if (!active) query = 0;
  const int n = query / L;
  const float* qrow = q + (size_t)query * D_MODEL;
  const int* irow = idx + (size_t)query * KNBR;
  const float* kbase = k + (size_t)n * S * D_MODEL;
  const float* vbase = v + (size_t)n * S * D_MODEL;

  #pragma unroll
  for (int t = lane; t < D_MODEL; t += 32) sq[wave][t] = qrow[t];
  const int i0 = irow[lane];
  const int i1 = irow[lane + 32];
  sidx[wave][lane] = i0;
  sidx[wave][lane + 32] = i1;
  __syncthreads();

  const float scale = 0.17677669529663688f;  // 1/sqrt(32)
  const float* k0r = kbase + (size_t)i0 * D_MODEL;
  const float* k1r = kbase + (size_t)i1 * D_MODEL;
  float s0[NHEAD], s1[NHEAD];
  #pragma unroll
  for (int h = 0; h < NHEAD; ++h) {
    float a0 = 0.f, a1 = 0.f;
    #pragma unroll
    for (int d = 0; d < HEAD_DIM; ++d) {
      const float qv = sq[wave][h * HEAD_DIM + d];
      a0 = fmaf(k0r[h * HEAD_DIM + d], qv, a0);
      a1 = fmaf(k1r[h * HEAD_DIM + d], qv, a1);
    }
    s0[h] = a0 * scale;
    s1[h] = a1 * scale;
  }
  #pragma unroll
  for (int h = 0; h < NHEAD; ++h) {
    float m = fmaxf(s0[h], s1[h]);
    for (int o = 16; o > 0; o >>= 1) m = fmaxf(m, __shfl_xor(m, o, 32));
    const float e0 = __expf(s0[h] - m);
    const float e1 = __expf(s1[h] - m);
    float s = e0 + e1;
    for (int o = 16; o > 0; o >>= 1) s += __shfl_xor(s, o, 32);
    const float inv = 1.f / s;
    sattn[wave][h * KNBR + lane] = e0 * inv;
    sattn[wave][h * KNBR + lane + 32] = e1 * inv;
  }
  __syncthreads();

  float acc[NHEAD] = {};
  for (int kk = 0; kk < KNBR; ++kk) {
    const float* vrow = vbase + (size_t)sidx[wave][kk] * D_MODEL;
    #pragma unroll
    for (int h = 0; h < NHEAD; ++h)
      acc[h] = fmaf(sattn[wave][h * KNBR + kk], vrow[h * HEAD_DIM + lane], acc[h]);
  }
  if (active) {
    float* mrow = msg + (size_t)query * D_MODEL;
    #pragma unroll
    for (int h = 0; h < NHEAD; ++h) mrow[h * HEAD_DIM + lane] = acc[h];
  }
}

// ---------------------------------------------------------------------------
// Block-wide (256 threads) sum reduction. red[] is 8 floats of LDS.
// ---------------------------------------------------------------------------
__device__ __forceinline__ float blockSum256(float v, float* red) {
  const int lane = threadIdx.x & 31;
  const int w = threadIdx.x >> 5;
  #pragma unroll
  for (int o = 16; o > 0; o >>= 1) v += __shfl_xor(v, o, 32);
  __syncthreads();  // protect red from any prior use
  if (lane == 0) red[w] = v;
  __syncthreads();
  float s = 0.f;
  #pragma unroll
  for (int i = 0; i < 8; ++i) s += red[i];
  return s;
}

// xm[row] = [ x[row] (256) | LN(t[row]; g,b) (256) ]
__global__ __launch_bounds__(256) void ln_concat(
    const float* __restrict__ t, const float* __restrict__ x,
    const float* __restrict__ g, const float* __restrict__ b,
    float* __restrict__ xm) {
  __shared__ float red[8];
  const int row = blockIdx.x, tid = threadIdx.x;
  const float v = t[(size_t)row * D_MODEL + tid];
  const float mean = blockSum256(v, red) * (1.f / D_MODEL);
  const float d = v - mean;
  const float var = blockSum256(d * d, red) * (1.f / D_MODEL);
  const float y = d * rsqrtf(var + LN_EPS) * g[tid] + b[tid];
  xm[(size_t)row * (2 * D_MODEL) + tid] = x[(size_t)row * D_MODEL + tid];
  xm[(size_t)row * (2 * D_MODEL) + D_MODEL + tid] = y;
}

// out[row] = x[row] + LN(t[row]; g,b)
__global__ __launch_bounds__(256) void residual_ln(
    const float* __restrict__ t, const float* __restrict__ x,
    const float* __restrict__ g, const float* __restrict__ b,
    float* __restrict__ out) {
  __shared__ float red[8];
  const int row = blockIdx.x, tid = threadIdx.x;
  const float v = t[(size_t)row * D_MODEL + tid];
  const float mean = blockSum256(v, red) * (1.f / D_MODEL);
  const float d = v - mean;
  const float var = blockSum256(d * d, red) * (1.f / D_MODEL);
  const float y = d * rsqrtf(var + LN_EPS) * g[tid] + b[tid];
  out[(size_t)row * D_MODEL + tid] = x[(size_t)row * D_MODEL + tid] + y;
}

extern "C" void kernel_launch(void* const* d_in, const int* in_sizes, int n_in,
                              void* d_out, int out_size, void* d_ws, size_t ws_size,
                              hipStream_t stream) {
  (void)n_in; (void)out_size; (void)ws_size;
  const float* x   = (const float*)d_in[0];
  const float* src = (const float*)d_in[1];
  const int*   eix = (const int*)d_in[2];
  const float* Wq  = (const float*)d_in[3];
  const float* Wk  = (const float*)d_in[4];
  const float* Wv  = (const float*)d_in[5];
  const float* Wm  = (const float*)d_in[6];
  const float* W1  = (const float*)d_in[7];
  const float* W2  = (const float*)d_in[8];
  const float* g1  = (const float*)d_in[9];
  const float* b1  = (const float*)d_in[10];
  const float* g2  = (const float*)d_in[11];
  const float* b2  = (const float*)d_in[12];

  const int NL = in_sizes[0] / D_MODEL;  // N*L query rows
  const int NS = in_sizes[1] / D_MODEL;  // N*S source rows
  const int NB = 2;                      // batch size per reference setup
  const int L = NL / NB;
  const int S = NS / NB;

  // Workspace layout (floats), with reuse:
  float* ws   = (float*)d_ws;
  float* q_   = ws;                         // NL*256
  float* k_   = q_ + (size_t)NL * D_MODEL;  // NS*256
  float* v_   = k_ + (size_t)NS * D_MODEL;  // NS*256
  float* msg_ = v_ + (size_t)NS * D_MODEL;  // NL*256
  float* h_   = msg_ + (size_t)NL * D_MODEL;// NL*512
  float* tmp_ = q_;                         // reuse q slot (q dead after attn)
  float* xm_  = k_;                         // reuse k+v slots (dead after attn)
  float* h2_  = q_;                         // reuse tmp slot (dead after concat)

  dim3 blk(256);
  const int mb = (NL + 127) / 128;
  const int sb = (NS + 127) / 128;

  // Q/K/V projections
  gemm_f32_wmma<<<dim3(D_MODEL / 64, mb), blk, 0, stream>>>(x,   Wq, q_, NL, D_MODEL, D_MODEL, 0);
  gemm_f32_wmma<<<dim3(D_MODEL / 64, sb), blk, 0, stream>>>(src, Wk, k_, NS, D_MODEL, D_MODEL, 0);
  gemm_f32_wmma<<<dim3(D_MODEL / 64, sb), blk, 0, stream>>>(src, Wv, v_, NS, D_MODEL, D_MODEL, 0);
  // Gathered epipolar attention -> msg
  epi_attn<<<(NL + 7) / 8, blk, 0, stream>>>(q_, k_, v_, eix, msg_, NL, L, S);
  // msg @ Wm -> tmp
  gemm_f32_wmma<<<dim3(D_MODEL / 64, mb), blk, 0, stream>>>(msg_, Wm, tmp_, NL, D_MODEL, D_MODEL, 0);
  // xm = [x | LN(tmp)]
  ln_concat<<<NL, blk, 0, stream>>>(tmp_, x, g1, b1, xm_);
  // h = relu(xm @ W1)
  gemm_f32_wmma<<<dim3((2 * D_MODEL) / 64, mb), blk, 0, stream>>>(xm_, W1, h_, NL, 2 * D_MODEL, 2 * D_MODEL, 1);
  // h2 = h @ W2
  gemm_f32_wmma<<<dim3(D_MODEL / 64, mb), blk, 0, stream>>>(h_, W2, h2_, NL, D_MODEL, 2 * D_MODEL, 0);
  // out = x + LN(h2)
  residual_ln<<<NL, blk, 0, stream>>>(h2_, x, g2, b2, (float*)d_out);
}